// CTM_70600672412340
// MI455X (gfx1250) — compile-verified
//
#include <hip/hip_runtime.h>
#include <math.h>

// Problem constants (match reference)
#define BB  16
#define NN  2048
#define CC  384
#define KK  512
#define KNN 5

typedef __attribute__((ext_vector_type(2))) float v2f;
typedef __attribute__((ext_vector_type(8))) float v8f;

// ---------------------------------------------------------------------------
// CDNA5 async global->LDS copy (ASYNCcnt-tracked, per-lane b128).
// Syntax: global_load_async_to_lds_b128 vdst(LDS byte addr), vaddr(64b), off
// Generic LDS pointers carry the wave-relative LDS byte address in the low
// 32 bits, so truncation yields the VDST value the instruction expects.
// ---------------------------------------------------------------------------
__device__ __forceinline__ void async_ld_b128(const float* gptr, const float* lptr) {
    unsigned loff = (unsigned)(unsigned long long)lptr;
    unsigned long long ga = (unsigned long long)gptr;
    asm volatile("global_load_async_to_lds_b128 %0, %1, off"
                 :: "v"(loff), "v"(ga) : "memory");
}
__device__ __forceinline__ void wait_async0() {
#if __has_builtin(__builtin_amdgcn_s_wait_asynccnt)
    __builtin_amdgcn_s_wait_asynccnt(0);
#else
    asm volatile("s_wait_asynccnt 0" ::: "memory");
#endif
}

// ---------------------------------------------------------------------------
// 1) Per-token prep: sq[i] = |x_i|^2 ; wtok[i] = exp(x_i . score_w + score_b)
// ---------------------------------------------------------------------------
__global__ void __launch_bounds__(256)
prep_kernel(const float* __restrict__ x, const float* __restrict__ sw,
            const float* __restrict__ sb, float* __restrict__ sq,
            float* __restrict__ wtok) {
    int wave = threadIdx.x >> 5, lane = threadIdx.x & 31;
    int token = blockIdx.x * 8 + wave;              // covers BB*NN tokens
    const float* xr = x + (size_t)token * CC;
    float s = 0.f, d = 0.f;
    for (int c = lane; c < CC; c += 32) { float v = xr[c]; s += v * v; d += v * sw[c]; }
    for (int off = 16; off > 0; off >>= 1) {
        s += __shfl_xor(s, off, 32);
        d += __shfl_xor(d, off, 32);
    }
    if (lane == 0) { sq[token] = s; wtok[token] = expf(d + sb[0]); }
}

__global__ void init_kernel(unsigned* __restrict__ distmax) {
    if (threadIdx.x < BB) distmax[threadIdx.x] = 0u;
}

// ---------------------------------------------------------------------------
// 2) WMMA fp32 GEMM + distance epilogue, with double-buffered async LDS
//    staging. WG = 256 threads = 8 waves; tile 128(M) x 64(N); wave w owns
//    rows [16w,16w+16) x 64 cols (4 accumulators of 16x16 f32).
//    dist[b][i][j] = sqrt(max(sq_i + sq_j - 2*G, 0)) / sqrt(C)
// ---------------------------------------------------------------------------
__global__ void __launch_bounds__(256)
gram_dist_kernel(const float* __restrict__ x, const float* __restrict__ sq,
                 float* __restrict__ dist) {
    __shared__ float As[2][128 * 16];   // 2 x 8 KB
    __shared__ float Bs[2][64 * 16];    // 2 x 4 KB
    const int b = blockIdx.z;
    const int rowBase = blockIdx.y * 128;
    const int colBase = blockIdx.x * 64;
    const float* xb  = x + (size_t)b * NN * CC;
    const float* sqb = sq + b * NN;
    float* db = dist + (size_t)b * NN * NN;
    const int tid  = threadIdx.x;
    const int wave = tid >> 5, lane = tid & 31;
    const int lhalf = lane >> 4, l16 = lane & 15;

    // async-stage one 16-wide K slice (A: 128x16, B: 64x16) into buffer bufi
    auto stage = [&](int bufi, int ko) {
        #pragma unroll
        for (int it = 0; it < 2; ++it) {
            int idx = tid + it * 256;               // 0..511 -> A tile float4s
            int r = idx >> 2, c4 = idx & 3;
            async_ld_b128(&xb[(size_t)(rowBase + r) * CC + ko + c4 * 4],
                          &As[bufi][r * 16 + c4 * 4]);
        }
        {
            int r = tid >> 2, c4 = tid & 3;         // 0..63 -> B tile float4s
            async_ld_b128(&xb[(size_t)(colBase + r) * CC + ko + c4 * 4],
                          &Bs[bufi][r * 16 + c4 * 4]);
        }
    };

    v8f acc[4] = {};

    stage(0, 0);
    wait_async0();
    __syncthreads();

    int buf = 0;
    for (int ko = 0; ko < CC; ko += 16) {
        if (ko + 16 < CC) stage(buf ^ 1, ko + 16);  // prefetch next slice (overlaps compute)

        #pragma unroll
        for (int ks = 0; ks < 4; ++ks) {
            // A frag 16x4 f32: lanes 0-15 hold K=0,1 ; lanes 16-31 hold K=2,3
            const int kbase = ks * 4 + lhalf * 2;
            v2f a = *(const v2f*)&As[buf][(wave * 16 + l16) * 16 + kbase];
            #pragma unroll
            for (int t = 0; t < 4; ++t) {
                v2f bf = *(const v2f*)&Bs[buf][(t * 16 + l16) * 16 + kbase];
                acc[t] = __builtin_amdgcn_wmma_f32_16x16x4_f32(
                    false, a, false, bf, (short)0, acc[t], false, false);
            }
        }

        wait_async0();       // own async prefetch landed in LDS
        __syncthreads();     // everyone's landed
        buf ^= 1;
    }

    const float inv = 0.05103103630798288f;  // 1/sqrt(384)
    #pragma unroll
    for (int t = 0; t < 4; ++t) {
        #pragma unroll
        for (int v = 0; v < 8; ++v) {
            // C/D layout: VGPR v -> M = v (lanes 0-15) or v+8 (lanes 16-31)
            int i = rowBase + wave * 16 + lhalf * 8 + v;
            int j = colBase + t * 16 + l16;
            float d2 = sqb[i] + sqb[j] - 2.0f * acc[t][v];
            // dist is 268 MB (> 192 MB L2): stream it out non-temporally
            __builtin_nontemporal_store(sqrtf(fmaxf(d2, 0.f)) * inv,
                                        &db[(size_t)i * NN + j]);
        }
    }
}

// ---------------------------------------------------------------------------
// 3) Per-row 5 smallest distances -> density; also per-batch max distance.
// ---------------------------------------------------------------------------
__device__ inline void insert5(float* t, float v) {
    if (v >= t[4]) return;
    t[4] = v;
    #pragma unroll
    for (int i = 4; i > 0; --i)
        if (t[i] < t[i - 1]) { float tmp = t[i]; t[i] = t[i - 1]; t[i - 1] = tmp; }
}
__device__ inline void merge5(float* a, const float* b) {
    float out[5]; int ia = 0, ib = 0;
    #pragma unroll
    for (int i = 0; i < 5; ++i) out[i] = (a[ia] <= b[ib]) ? a[ia++] : b[ib++];
    #pragma unroll
    for (int i = 0; i < 5; ++i) a[i] = out[i];
}

__global__ void __launch_bounds__(256)
density_kernel(const float* __restrict__ dist, const float* __restrict__ noise,
               float* __restrict__ density, unsigned* __restrict__ distmax) {
    __shared__ float st[256 * 5];
    __shared__ float sm[256];
    int row = blockIdx.x;                     // b*NN + i
    int b = row / NN;
    const float* dr = dist + (size_t)row * NN;
    float t5[5] = {1e30f, 1e30f, 1e30f, 1e30f, 1e30f};
    float mx = 0.f;
    for (int j = threadIdx.x; j < NN; j += 256) {
        float v = __builtin_nontemporal_load(&dr[j]);   // streaming pass
        insert5(t5, v);
        mx = fmaxf(mx, v);
    }
    for (int i = 0; i < 5; ++i) st[threadIdx.x * 5 + i] = t5[i];
    sm[threadIdx.x] = mx;
    __syncthreads();
    for (int off = 128; off > 0; off >>= 1) {
        if (threadIdx.x < off) {
            float mine[5], other[5];
            for (int i = 0; i < 5; ++i) {
                mine[i]  = st[threadIdx.x * 5 + i];
                other[i] = st[(threadIdx.x + off) * 5 + i];
            }
            merge5(mine, other);
            for (int i = 0; i < 5; ++i) st[threadIdx.x * 5 + i] = mine[i];
            sm[threadIdx.x] = fmaxf(sm[threadIdx.x], sm[threadIdx.x + off]);
        }
        __syncthreads();
    }
    if (threadIdx.x == 0) {
        float s = 0.f;
        for (int i = 0; i < 5; ++i) s += st[i] * st[i];
        density[row] = expf(-s * (1.0f / KNN)) + noise[row] * 1e-6f;
        atomicMax(&distmax[b], __float_as_uint(sm[0]));  // dist >= 0: uint order == float order
    }
}

// ---------------------------------------------------------------------------
// 4) dmin over higher-density peers (capped by dist_max) -> score
// ---------------------------------------------------------------------------
__global__ void __launch_bounds__(256)
score_kernel(const float* __restrict__ dist, const float* __restrict__ density,
             const unsigned* __restrict__ distmax, float* __restrict__ score) {
    __shared__ float red[256];
    int row = blockIdx.x;
    int b = row / NN;
    const float* dr  = dist + (size_t)row * NN;
    const float* den = density + b * NN;
    float di = density[row];
    float m = __uint_as_float(distmax[b]);
    for (int j = threadIdx.x; j < NN; j += 256) {
        float dj = __builtin_nontemporal_load(&dr[j]);
        if (den[j] > di) m = fminf(m, dj);
    }
    red[threadIdx.x] = m;
    __syncthreads();
    for (int off = 128; off > 0; off >>= 1) {
        if (threadIdx.x < off) red[threadIdx.x] = fminf(red[threadIdx.x], red[threadIdx.x + off]);
        __syncthreads();
    }
    if (threadIdx.x == 0) score[row] = red[0] * di;
}

// ---------------------------------------------------------------------------
// 5) Deterministic top-512 by rank counting (stable, matches lax.top_k ties)
// ---------------------------------------------------------------------------
__global__ void __launch_bounds__(256)
rank_kernel(const float* __restrict__ score, int* __restrict__ index_down) {
    __shared__ float ss[NN];
    int g = blockIdx.x * 256 + threadIdx.x;
    int b = g / NN, i = g % NN;
    const float* sc = score + b * NN;
    for (int j = threadIdx.x; j < NN; j += 256) ss[j] = sc[j];
    __syncthreads();
    float si = ss[i];
    int rank = 0;
    for (int j = 0; j < NN; ++j) {
        float sj = ss[j];
        rank += (sj > si) || (sj == si && j < i);
    }
    if (rank < KK) index_down[b * KK + rank] = i;
}

// ---------------------------------------------------------------------------
// 6) Assign each token to nearest center (first-min tie-break = argmin)
// ---------------------------------------------------------------------------
__global__ void __launch_bounds__(256)
assign_kernel(const float* __restrict__ dist, const int* __restrict__ index_down,
              int* __restrict__ idx_cluster) {
    __shared__ int ctr[KK];
    int b = blockIdx.y;
    int n = blockIdx.x * 256 + threadIdx.x;
    for (int k = threadIdx.x; k < KK; k += 256) ctr[k] = index_down[b * KK + k];
    __syncthreads();
    const float* db = dist + (size_t)b * NN * NN;
    float best = 1e30f; int bk = 0;
    for (int k = 0; k < KK; ++k) {
        float d = db[(size_t)ctr[k] * NN + n];   // coalesced over n per k
        if (d < best) { best = d; bk = k; }
    }
    idx_cluster[b * NN + n] = bk;
}

__global__ void override_kernel(const int* __restrict__ index_down,
                                int* __restrict__ idx_cluster) {
    int g = blockIdx.x * 256 + threadIdx.x;      // BB*KK
    int b = g / KK, k = g % KK;
    idx_cluster[b * NN + index_down[g]] = k;
}

// ---------------------------------------------------------------------------
// 7) Deterministic counting-sort by cluster: pos/count/scan/scatter
// ---------------------------------------------------------------------------
__global__ void __launch_bounds__(256)
pos_kernel(const int* __restrict__ idx_cluster, int* __restrict__ posArr) {
    __shared__ int sc[NN];
    int g = blockIdx.x * 256 + threadIdx.x;
    int b = g / NN, n = g % NN;
    const int* ic = idx_cluster + b * NN;
    for (int j = threadIdx.x; j < NN; j += 256) sc[j] = ic[j];
    __syncthreads();
    int me = sc[n], p = 0;
    for (int m = 0; m < n; ++m) p += (sc[m] == me);
    posArr[g] = p;
}

__global__ void __launch_bounds__(256)
count_kernel(const int* __restrict__ idx_cluster, int* __restrict__ cnt) {
    __shared__ int sc[NN];
    int g = blockIdx.x * 256 + threadIdx.x;      // BB*KK ids; block spans one batch
    int b = g / KK, k = g % KK;
    const int* ic = idx_cluster + b * NN;
    for (int j = threadIdx.x; j < NN; j += 256) sc[j] = ic[j];
    __syncthreads();
    int c = 0;
    for (int m = 0; m < NN; ++m) c += (sc[m] == k);
    cnt[g] = c;
}

__global__ void scan_kernel(const int* __restrict__ cnt, int* __restrict__ startoff) {
    int b = blockIdx.x;
    if (threadIdx.x == 0) {
        int acc = 0;
        for (int k = 0; k < KK; ++k) { startoff[b * KK + k] = acc; acc += cnt[b * KK + k]; }
    }
}

__global__ void __launch_bounds__(256)
scatter_kernel(const int* __restrict__ idx_cluster, const int* __restrict__ posArr,
               const int* __restrict__ startoff, int* __restrict__ members) {
    int g = blockIdx.x * 256 + threadIdx.x;
    int b = g / NN, n = g % NN;
    int cl = idx_cluster[g];
    members[b * NN + startoff[b * KK + cl] + posArr[g]] = n;
}

// ---------------------------------------------------------------------------
// 8) Merge tokens per cluster (deterministic ordered sums), write x_merged
// ---------------------------------------------------------------------------
__global__ void __launch_bounds__(384)
merge_kernel(const float* __restrict__ x, const float* __restrict__ wtok,
             const int* __restrict__ members, const int* __restrict__ cnt,
             const int* __restrict__ startoff, float* __restrict__ normw,
             float* __restrict__ xm) {
    __shared__ int   ml[NN];   // 8 KB
    __shared__ float wl[NN];   // 8 KB
    __shared__ float sAll;
    int g = blockIdx.x;        // b*KK + k
    int b = g / KK;
    int cntk  = cnt[g];
    int start = startoff[g];
    const int* mem = members + b * NN + start;
    for (int m = threadIdx.x; m < cntk; m += blockDim.x) ml[m] = mem[m];
    __syncthreads();
    if (threadIdx.x == 0) {                       // ordered serial sum: deterministic
        float s = 0.f;
        for (int m = 0; m < cntk; ++m) s += wtok[b * NN + ml[m]];
        sAll = s + 1e-6f;
    }
    __syncthreads();
    float s = sAll;
    for (int m = threadIdx.x; m < cntk; m += blockDim.x) {
        float nw = wtok[b * NN + ml[m]] / s;
        wl[m] = nw;
        normw[b * NN + ml[m]] = nw;
    }
    __syncthreads();
    const float* xb = x + (size_t)b * NN * CC;
    int ch = threadIdx.x;                         // exactly CC threads
    float acc = 0.f;
    for (int m = 0; m < cntk; ++m)
        acc += xb[(size_t)ml[m] * CC + ch] * wl[m];
    xm[(size_t)g * CC + ch] = acc;
}

// ---------------------------------------------------------------------------
// 9) Final gathers: idx_token_new and agg_weight_new
// ---------------------------------------------------------------------------
__global__ void __launch_bounds__(256)
final_kernel(const int* __restrict__ idx_token, const float* __restrict__ agg,
             const int* __restrict__ idx_cluster, const float* __restrict__ normw,
             float* __restrict__ out_idx, float* __restrict__ out_agg) {
    int g = blockIdx.x * 256 + threadIdx.x;
    int b = g / NN;
    int t = idx_token[g];
    int cl = idx_cluster[b * NN + t];
    out_idx[g] = (float)cl;
    out_agg[g] = agg[g] * normw[b * NN + t];
}

// ---------------------------------------------------------------------------
extern "C" void kernel_launch(void* const* d_in, const int* in_sizes, int n_in,
                              void* d_out, int out_size, void* d_ws, size_t ws_size,
                              hipStream_t stream) {
    const float* x        = (const float*)d_in[0];
    const int*   idx_tok  = (const int*)d_in[1];
    const float* agg      = (const float*)d_in[2];
    const float* noise    = (const float*)d_in[3];
    const float* sw       = (const float*)d_in[4];
    const float* sb       = (const float*)d_in[5];

    char* ws = (char*)d_ws;
    size_t off = 0;
    auto alloc = [&](size_t bytes) -> void* {
        void* p = ws + off;
        off += (bytes + 255) & ~(size_t)255;
        return p;
    };
    float*    dist       = (float*)   alloc((size_t)BB * NN * NN * 4);   // 268 MB
    float*    sq         = (float*)   alloc((size_t)BB * NN * 4);
    float*    wtok       = (float*)   alloc((size_t)BB * NN * 4);
    float*    density    = (float*)   alloc((size_t)BB * NN * 4);
    float*    score      = (float*)   alloc((size_t)BB * NN * 4);
    unsigned* distmax    = (unsigned*)alloc((size_t)BB * 4);
    int*      index_down = (int*)     alloc((size_t)BB * KK * 4);
    int*      idx_clu    = (int*)     alloc((size_t)BB * NN * 4);
    int*      cnt        = (int*)     alloc((size_t)BB * KK * 4);
    int*      startoff   = (int*)     alloc((size_t)BB * KK * 4);
    int*      posArr     = (int*)     alloc((size_t)BB * NN * 4);
    int*      members    = (int*)     alloc((size_t)BB * NN * 4);
    float*    normw      = (float*)   alloc((size_t)BB * NN * 4);

    float* xm      = (float*)d_out;                         // [B,K,C]
    float* out_idx = xm + (size_t)BB * KK * CC;             // [B,N] (as float)
    float* out_agg = out_idx + (size_t)BB * NN;             // [B,N]

    prep_kernel     <<<BB * NN / 8, 256, 0, stream>>>(x, sw, sb, sq, wtok);
    init_kernel     <<<1, 256, 0, stream>>>(distmax);
    gram_dist_kernel<<<dim3(NN / 64, NN / 128, BB), 256, 0, stream>>>(x, sq, dist);
    density_kernel  <<<BB * NN, 256, 0, stream>>>(dist, noise, density, distmax);
    score_kernel    <<<BB * NN, 256, 0, stream>>>(dist, density, distmax, score);
    rank_kernel     <<<BB * NN / 256, 256, 0, stream>>>(score, index_down);
    assign_kernel   <<<dim3(NN / 256, BB), 256, 0, stream>>>(dist, index_down, idx_clu);
    override_kernel <<<BB * KK / 256, 256, 0, stream>>>(index_down, idx_clu);
    pos_kernel      <<<BB * NN / 256, 256, 0, stream>>>(idx_clu, posArr);
    count_kernel    <<<BB * KK / 256, 256, 0, stream>>>(idx_clu, cnt);
    scan_kernel     <<<BB, 32, 0, stream>>>(cnt, startoff);
    scatter_kernel  <<<BB * NN / 256, 256, 0, stream>>>(idx_clu, posArr, startoff, members);
    merge_kernel    <<<BB * KK, 384, 0, stream>>>(x, wtok, members, cnt, startoff, normw, xm);
    final_kernel    <<<BB * NN / 256, 256, 0, stream>>>(idx_tok, agg, idx_clu, normw,
                                                        out_idx, out_agg);
}